// SpatioConvLayer_81836306858001
// MI455X (gfx1250) — compile-verified
//
#include <hip/hip_runtime.h>
#include <math.h>

typedef __attribute__((ext_vector_type(2))) float v2f;
typedef __attribute__((ext_vector_type(8))) float v8f;

#define C_CH 96   // C_IN == C_OUT == 96

// ---------------------------------------------------------------------------
// 1) init: deg = 1.0 (self loop), acc = 0
// ---------------------------------------------------------------------------
__global__ void init_kernel(float* __restrict__ deg, float* __restrict__ acc,
                            int n, int total_nc) {
    int i = blockIdx.x * blockDim.x + threadIdx.x;
    if (i < total_nc) acc[i] = 0.0f;
    if (i < n)        deg[i] = 1.0f;
}

// ---------------------------------------------------------------------------
// 2) degree accumulation over target nodes
// ---------------------------------------------------------------------------
__global__ void degree_kernel(const int* __restrict__ colIdx,
                              float* __restrict__ deg, int E) {
    int e = blockIdx.x * blockDim.x + threadIdx.x;
    if (e < E) atomicAdd(&deg[colIdx[e]], 1.0f);
}

// ---------------------------------------------------------------------------
// 3) dinv = rsqrt(deg) in place (deg >= 1 always, keep reference guard)
// ---------------------------------------------------------------------------
__global__ void rsqrt_kernel(float* __restrict__ deg, int n) {
    int i = blockIdx.x * blockDim.x + threadIdx.x;
    if (i < n) {
        float d = deg[i];
        deg[i] = (d > 0.0f) ? rsqrtf(d) : 0.0f;
    }
}

// ---------------------------------------------------------------------------
// 4) h = x @ W via V_WMMA_F32_16X16X4_F32, one wave per 16x16 output tile.
//    A (16x4 f32): lanes 0-15 -> {K, K+1}, lanes 16-31 -> {K+2, K+3}
//    B (4x16 f32): same K split across lane halves, N = lane & 15
//    C/D (16x16 f32): VGPR r: lanes 0-15 -> M=r, lanes 16-31 -> M=r+8
// ---------------------------------------------------------------------------
__global__ __launch_bounds__(32) void gemm_wmma_kernel(
        const float* __restrict__ x, const float* __restrict__ W,
        float* __restrict__ h, int n) {
    const int lane = threadIdx.x;
    const int row0 = blockIdx.x * 16;
    const int col0 = blockIdx.y * 16;
    const int mn   = lane & 15;          // M for A-frag, N for B/C-frag
    const int koff = (lane >> 4) << 1;   // 0 for lanes 0-15, 2 for 16-31

    // clamp row reads so EXEC stays all-ones through the WMMA loop
    int arow = row0 + mn; if (arow >= n) arow = n - 1;
    const float* xrow = x + (size_t)arow * C_CH;

    v8f c = {};
#pragma unroll
    for (int k = 0; k < C_CH; k += 4) {
        v2f a, b;
        a.x = xrow[k + koff];
        a.y = xrow[k + koff + 1];
        b.x = W[(size_t)(k + koff)     * C_CH + col0 + mn];
        b.y = W[(size_t)(k + koff + 1) * C_CH + col0 + mn];
        // 8 args: (neg_a, A, neg_b, B, c_mod, C, reuse_a, reuse_b)
        c = __builtin_amdgcn_wmma_f32_16x16x4_f32(
                false, a, false, b, (short)0, c, false, false);
    }

    const int mbase = (lane >> 4) * 8;
#pragma unroll
    for (int r = 0; r < 8; ++r) {
        int m = row0 + mbase + r;
        if (m < n) h[(size_t)m * C_CH + col0 + mn] = c[r];
    }
}

// ---------------------------------------------------------------------------
// 5) scatter: one thread per (edge, 4-channel chunk); float4 gather of h[row],
//    scale by dinv[row]*dinv[col], atomic-add into acc[col]
// ---------------------------------------------------------------------------
__global__ void scatter_kernel(const int* __restrict__ rowIdx,
                               const int* __restrict__ colIdx,
                               const float* __restrict__ h,
                               const float* __restrict__ dinv,
                               float* __restrict__ acc, int E) {
    long long idx = (long long)blockIdx.x * blockDim.x + threadIdx.x;
    int e  = (int)(idx / (C_CH / 4));
    int c4 = (int)(idx % (C_CH / 4)) * 4;
    if (e >= E) return;

    int r = rowIdx[e];
    int c = colIdx[e];
    float norm = dinv[r] * dinv[c];

    const float4 v = *reinterpret_cast<const float4*>(h + (size_t)r * C_CH + c4);
    float* ap = acc + (size_t)c * C_CH + c4;
    atomicAdd(ap + 0, v.x * norm);
    atomicAdd(ap + 1, v.y * norm);
    atomicAdd(ap + 2, v.z * norm);
    atomicAdd(ap + 3, v.w * norm);
}

// ---------------------------------------------------------------------------
// 6) finalize: out = prelu(acc + h*dinv^2 (self loop) + b)
// ---------------------------------------------------------------------------
__global__ void finalize_kernel(const float* __restrict__ acc,
                                const float* __restrict__ h,
                                const float* __restrict__ dinv,
                                const float* __restrict__ bias,
                                const float* __restrict__ prelu_a,
                                float* __restrict__ out, int total_nc) {
    int idx = blockIdx.x * blockDim.x + threadIdx.x;
    if (idx >= total_nc) return;
    int i = idx / C_CH;
    int c = idx % C_CH;
    float di = dinv[i];
    float v  = acc[idx] + h[idx] * (di * di) + bias[c];
    float a  = prelu_a[0];
    out[idx] = (v >= 0.0f) ? v : a * v;
}

// ---------------------------------------------------------------------------
extern "C" void kernel_launch(void* const* d_in, const int* in_sizes, int n_in,
                              void* d_out, int out_size, void* d_ws, size_t ws_size,
                              hipStream_t stream) {
    const float* x  = (const float*)d_in[0];
    const float* W  = (const float*)d_in[1];
    const float* b  = (const float*)d_in[2];
    const float* pa = (const float*)d_in[3];
    const int*   ei = (const int*)d_in[4];

    const int N = in_sizes[0] / C_CH;
    const int E = in_sizes[4] / 2;
    const int* rowIdx = ei;       // edge_index[0] : sources
    const int* colIdx = ei + E;   // edge_index[1] : targets

    // workspace layout: [deg/dinv: N] [h: N*96] [acc: N*96]
    float* deg = (float*)d_ws;
    float* h   = deg + N;
    float* acc = h + (size_t)N * C_CH;

    const int total_nc = N * C_CH;

    init_kernel<<<(total_nc + 255) / 256, 256, 0, stream>>>(deg, acc, N, total_nc);
    degree_kernel<<<(E + 255) / 256, 256, 0, stream>>>(colIdx, deg, E);
    rsqrt_kernel<<<(N + 255) / 256, 256, 0, stream>>>(deg, N);

    dim3 gemm_grid((N + 15) / 16, C_CH / 16);
    gemm_wmma_kernel<<<gemm_grid, 32, 0, stream>>>(x, W, h, N);

    long long scat_threads = (long long)E * (C_CH / 4);
    scatter_kernel<<<(unsigned)((scat_threads + 255) / 256), 256, 0, stream>>>(
        rowIdx, colIdx, h, deg, acc, E);

    finalize_kernel<<<(total_nc + 255) / 256, 256, 0, stream>>>(
        acc, h, deg, b, pa, (float*)d_out, total_nc);
}